// LSTMPredictor_27101243638312
// MI455X (gfx1250) — compile-verified
//
#include <hip/hip_runtime.h>
#include <math.h>

// ---------------------------------------------------------------------------
// LSTM (T=8192, IN=H=512) for MI455X (gfx1250, wave32).
//
// Phase 1: GX[t][0:2048] = x_t @ W_ih^T + (b_ih + b_hh)   -- parallel GEMM
//          using V_WMMA_F32_16X16X4_F32. Each wave computes 4 N-tiles per
//          A fragment (register reuse: 5 b64 loads per 4 WMMAs).
// Phase 2: sequential scan with 32 persistent workgroups, W_hh slices served
//          from L2 (4MB total), one device-scope sense-reversal barrier per
//          timestep, double-buffered h, prefetch of next step's gate slice.
// ---------------------------------------------------------------------------

#define T_LEN 8192
#define IN_D  512
#define H_D   512
#define G4    2048          // 4*H
#define NWG   32            // workgroups in the scan phase
#define HB    16            // hidden units per workgroup (NWG*HB == H)

typedef float v2f __attribute__((ext_vector_type(2)));
typedef float v8f __attribute__((ext_vector_type(8)));

// ---------------------------------------------------------------------------
// Phase 1: GX = X @ W_ih^T + bias, fp32 WMMA.
// Grid: 512 t-blocks x 4 n-blocks = 2048 blocks, 256 threads (8 waves).
// Block covers 16 rows x 512 cols. Wave w owns cols nb*512 + w*16 + j*128,
// j = 0..3: one A fragment feeds 4 WMMAs per K-step.
// A (16x4 f32): lanes 0-15 hold K={0,1}, lanes 16-31 hold K={2,3} (2 VGPRs).
// B (4x16 f32): mirrored; lane n holds W_ih[c_j+n][k..k+1] per half.
// C/D (16x16 f32): VGPR v -> M = half*8 + v, N = lane&15.
// ---------------------------------------------------------------------------
__launch_bounds__(256)
__global__ void gemm_gx_wmma(const float* __restrict__ X,
                             const float* __restrict__ Wih,
                             const float* __restrict__ bih,
                             const float* __restrict__ bhh,
                             float* __restrict__ GX) {
  const int wave = threadIdx.x >> 5;
  const int lane = threadIdx.x & 31;
  const int nb   = blockIdx.x & 3;         // 0..3   (512-col slabs)
  const int tb   = blockIdx.x >> 2;        // 0..511 (16-row slabs)
  const int t0   = tb * 16;
  const int c0   = nb * 512 + wave * 16;   // first of this wave's 4 tiles
  const int half = lane >> 4;              // 0: K pair {0,1}; 1: K pair {2,3}
  const int l    = lane & 15;

  const float* arow = X + (size_t)(t0 + l) * IN_D + 2 * half;      // M = t0+l
  const float* b0 = Wih + (size_t)(c0 +   0 + l) * IN_D + 2 * half;
  const float* b1 = Wih + (size_t)(c0 + 128 + l) * IN_D + 2 * half;
  const float* b2 = Wih + (size_t)(c0 + 256 + l) * IN_D + 2 * half;
  const float* b3 = Wih + (size_t)(c0 + 384 + l) * IN_D + 2 * half;

  v8f acc0 = {}, acc1 = {}, acc2 = {}, acc3 = {};
#pragma unroll 4
  for (int k = 0; k < IN_D; k += 4) {
    const v2f a  = *(const v2f*)(arow + k);
    const v2f w0 = *(const v2f*)(b0 + k);
    const v2f w1 = *(const v2f*)(b1 + k);
    const v2f w2 = *(const v2f*)(b2 + k);
    const v2f w3 = *(const v2f*)(b3 + k);
    acc0 = __builtin_amdgcn_wmma_f32_16x16x4_f32(false, a, false, w0,
                                                 (short)0, acc0, false, false);
    acc1 = __builtin_amdgcn_wmma_f32_16x16x4_f32(false, a, false, w1,
                                                 (short)0, acc1, false, false);
    acc2 = __builtin_amdgcn_wmma_f32_16x16x4_f32(false, a, false, w2,
                                                 (short)0, acc2, false, false);
    acc3 = __builtin_amdgcn_wmma_f32_16x16x4_f32(false, a, false, w3,
                                                 (short)0, acc3, false, false);
  }

#pragma unroll
  for (int j = 0; j < 4; ++j) {
    const v8f acc = (j == 0) ? acc0 : (j == 1) ? acc1 : (j == 2) ? acc2 : acc3;
    const int col = c0 + j * 128 + l;
    const float bias = bih[col] + bhh[col];
#pragma unroll
    for (int v = 0; v < 8; ++v) {
      const int m = half * 8 + v;
      GX[(size_t)(t0 + m) * G4 + col] = acc[v] + bias;
    }
  }
}

// ---------------------------------------------------------------------------
// Workspace init (d_ws is poisoned 0xAA by the harness).
// ---------------------------------------------------------------------------
__global__ void init_ws(float* __restrict__ h_buf, int* __restrict__ bar) {
  const int i = threadIdx.x;
  if (i < 2 * H_D) h_buf[i] = 0.0f;   // h0 = 0
  if (i < 2)       bar[i]   = 0;      // barrier count, phase
}

// ---------------------------------------------------------------------------
// Phase 2: sequential LSTM scan. 32 persistent WGs x 256 threads.
// WG `wg` owns hidden units [wg*16, wg*16+16): 64 rows of W_hh (i,f,g,o).
// Per step: copy h -> LDS, 64 L2-resident row dots (8 waves x 8 rows, shuffle
// reduction), gate math on 16 lanes, publish h slice, device barrier.
// ---------------------------------------------------------------------------
__launch_bounds__(256, 1)
__global__ void lstm_scan(const float* __restrict__ GX,
                          const float* __restrict__ Whh,
                          const float* __restrict__ likes,
                          float* __restrict__ h_buf,   // 2 * 512 (double buffer)
                          int* __restrict__ bar,       // [count, phase]
                          float* __restrict__ out) {   // 512
  __shared__ float h_lds[H_D];
  __shared__ float dots[64];
  __shared__ float c_lds[HB];

  const int tid  = threadIdx.x;
  const int wave = tid >> 5;
  const int lane = tid & 31;
  const int wg   = blockIdx.x;
  const int j0   = wg * HB;

  if (tid < HB) c_lds[tid] = 0.0f;
  int sense = 0;

  for (int t = 0; t < T_LEN; ++t) {
    // ---- prefetch next step's gate slice + likes row (global_prefetch) ---
    if (t + 1 < T_LEN && lane == 0) {
      if (wave < 4)
        __builtin_prefetch(GX + (size_t)(t + 1) * G4 + wave * H_D + j0, 0, 0);
      if (wave == 4 && t + 1 < T_LEN - 1)
        __builtin_prefetch(likes + (size_t)(t + 1) * H_D + j0, 0, 0);
    }

    // ---- load current h (written by all WGs last step) into LDS ----------
    const float* hsrc = h_buf + (t & 1) * H_D;
    for (int i = tid; i < H_D; i += 256) {
      h_lds[i] = __hip_atomic_load(&hsrc[i], __ATOMIC_RELAXED,
                                   __HIP_MEMORY_SCOPE_AGENT);
    }
    __syncthreads();

    // ---- 64 row dots: wave handles rows [wave*8, wave*8+8) ---------------
#pragma unroll
    for (int r8 = 0; r8 < 8; ++r8) {
      const int r = wave * 8 + r8;          // local row: g*16 + u
      const int g = r >> 4, u = r & 15;
      const float* wr = Whh + (size_t)(g * H_D + j0 + u) * H_D;
      float s = 0.0f;
#pragma unroll
      for (int i = 0; i < 4; ++i) {
        const int k = i * 128 + lane * 4;
        const float4 w4 = *(const float4*)(wr + k);
        const float4 h4 = *(const float4*)(&h_lds[k]);
        s += w4.x * h4.x + w4.y * h4.y + w4.z * h4.z + w4.w * h4.w;
      }
#pragma unroll
      for (int off = 16; off; off >>= 1) s += __shfl_xor(s, off, 32);
      if (lane == 0) dots[r] = s;
    }
    __syncthreads();

    // ---- gate math for this WG's 16 hidden units -------------------------
    if (tid < HB) {
      const int j = j0 + tid;
      const float* gxt = GX + (size_t)t * G4;
      const float gi = gxt[j]            + dots[0 * HB + tid];
      const float gf = gxt[H_D + j]      + dots[1 * HB + tid];
      const float gg = gxt[2 * H_D + j]  + dots[2 * HB + tid];
      const float go = gxt[3 * H_D + j]  + dots[3 * HB + tid];
      const float i_ = 1.0f / (1.0f + __expf(-gi));
      const float f_ = 1.0f / (1.0f + __expf(-gf));
      const float g_ = tanhf(gg);
      const float o_ = 1.0f / (1.0f + __expf(-go));
      const float c  = f_ * c_lds[tid] + i_ * g_;
      c_lds[tid] = c;
      const float lk = (t < T_LEN - 1) ? likes[(size_t)t * H_D + j] : 0.0f;
      const float h  = o_ * tanhf(c) + lk;
      __hip_atomic_store(&h_buf[((t + 1) & 1) * H_D + j], h,
                         __ATOMIC_RELAXED, __HIP_MEMORY_SCOPE_AGENT);
      if (t == T_LEN - 1) out[j] = h;
    }

    // ---- device-wide sense-reversal barrier (one per step) ---------------
    __threadfence();
    __syncthreads();
    if (tid == 0) {
      sense ^= 1;
      const int old = __hip_atomic_fetch_add(&bar[0], 1, __ATOMIC_ACQ_REL,
                                             __HIP_MEMORY_SCOPE_AGENT);
      if (old == NWG - 1) {
        __hip_atomic_store(&bar[0], 0, __ATOMIC_RELAXED,
                           __HIP_MEMORY_SCOPE_AGENT);
        __hip_atomic_store(&bar[1], sense, __ATOMIC_RELEASE,
                           __HIP_MEMORY_SCOPE_AGENT);
      } else {
        while (__hip_atomic_load(&bar[1], __ATOMIC_ACQUIRE,
                                 __HIP_MEMORY_SCOPE_AGENT) != sense) {
          __builtin_amdgcn_s_sleep(1);
        }
      }
    }
    __syncthreads();
  }
}

// ---------------------------------------------------------------------------
extern "C" void kernel_launch(void* const* d_in, const int* in_sizes, int n_in,
                              void* d_out, int out_size, void* d_ws,
                              size_t ws_size, hipStream_t stream) {
  (void)in_sizes; (void)n_in; (void)out_size; (void)ws_size;

  const float* x     = (const float*)d_in[0];  // (T, IN)
  const float* likes = (const float*)d_in[1];  // (T-1, H)
  const float* w_ih  = (const float*)d_in[2];  // (4H, IN)
  const float* w_hh  = (const float*)d_in[3];  // (4H, H)
  const float* b_ih  = (const float*)d_in[4];  // (4H,)
  const float* b_hh  = (const float*)d_in[5];  // (4H,)
  float* out = (float*)d_out;                  // (H,)

  // Workspace layout: GX (T*4H f32 = 67MB) | h double-buffer | barrier ints
  float* gx    = (float*)d_ws;
  float* h_buf = gx + (size_t)T_LEN * G4;
  int*   bar   = (int*)(h_buf + 2 * H_D);

  init_ws<<<1, 1024, 0, stream>>>(h_buf, bar);

  // 512 t-blocks * 4 n-blocks; 8 waves/block, 4 16x16 WMMA tiles per wave.
  gemm_gx_wmma<<<dim3(512 * 4), dim3(256), 0, stream>>>(x, w_ih, b_ih, b_hh,
                                                        gx);

  lstm_scan<<<dim3(NWG), dim3(256), 0, stream>>>(gx, w_hh, likes, h_buf, bar,
                                                 out);
}